// COGNet_80126909874537
// MI455X (gfx1250) — compile-verified
//
#include <hip/hip_runtime.h>
#include <math.h>

#define Bsz   512
#define Tv    16
#define Lc    32
#define Hvis  15
#define Nh    1024
#define Dd    64
#define NTOK  152
#define NMED  150
#define MAXLEN 45

typedef float v2f __attribute__((ext_vector_type(2)));
typedef float v8f __attribute__((ext_vector_type(8)));

// D(16x16) += A(16x64) * B(64x16), fp32 WMMA 16x16x4, A row-major (lda), B row-major (ldb)
__device__ __forceinline__ v8f wmma16x16_k64(const float* A, int lda,
                                             const float* B, int ldb,
                                             v8f acc, int lane)
{
  const int half = lane >> 4;   // 0: K pair {k,k+1}, 1: {k+2,k+3}
  const int l16  = lane & 15;   // A row / B col
#pragma unroll
  for (int k = 0; k < 64; k += 4) {
    v2f a, b;
    a.x = A[l16 * lda + k + 2 * half + 0];
    a.y = A[l16 * lda + k + 2 * half + 1];
    b.x = B[(k + 2 * half + 0) * ldb + l16];
    b.y = B[(k + 2 * half + 1) * ldb + l16];
    acc = __builtin_amdgcn_wmma_f32_16x16x4_f32(false, a, false, b, (short)0, acc,
                                                false, false);
  }
  return acc;
}

// Same but B given transposed: B[k][n] = Bt[n*ldb + k]  (torch weight layout W[n][k])
__device__ __forceinline__ v8f wmma16x16_k64_bt(const float* A, int lda,
                                                const float* Bt, int ldb,
                                                v8f acc, int lane)
{
  const int half = lane >> 4;
  const int l16  = lane & 15;
#pragma unroll
  for (int k = 0; k < 64; k += 4) {
    v2f a, b;
    a.x = A[l16 * lda + k + 2 * half + 0];
    a.y = A[l16 * lda + k + 2 * half + 1];
    b.x = Bt[l16 * ldb + k + 2 * half + 0];
    b.y = Bt[l16 * ldb + k + 2 * half + 1];
    acc = __builtin_amdgcn_wmma_f32_16x16x4_f32(false, a, false, b, (short)0, acc,
                                                false, false);
  }
  return acc;
}

// ---------------- encoder: per (b,t) diag attention ----------------
__global__ __launch_bounds__(128)
void k_enc(const int* __restrict__ diag_ids, const unsigned char* __restrict__ diag_mask,
           const float* __restrict__ diag_emb, const float* __restrict__ W1,
           const float* __restrict__ b1, const float* __restrict__ w2,
           const float* __restrict__ b2, float* __restrict__ v_all)
{
  __shared__ float Els[Lc][65];
  __shared__ float Wls[Dd][65];
  __shared__ float Gls[Lc][65];
  __shared__ float Sl[Lc];
  __shared__ float alpha[Lc];
  __shared__ unsigned char mk[Lc];
  const int bt  = blockIdx.x;         // b*T + t
  const int tid = threadIdx.x;

  for (int i = tid; i < Lc * Dd; i += 128) {
    int l = i >> 6, d = i & 63;
    int id = diag_ids[bt * Lc + l];
    Els[l][d] = (id != 0) ? diag_emb[id * Dd + d] : 0.0f;
  }
  for (int i = tid; i < Dd * Dd; i += 128) Wls[i >> 6][i & 63] = W1[i];
  if (tid < Lc) mk[tid] = diag_mask[bt * Lc + tid];
  __syncthreads();

  const int wave = tid >> 5, lane = tid & 31;
  const int half = lane >> 4, l16 = lane & 15;
  for (int mt = 0; mt < 2; mt++) {
    v8f acc;
#pragma unroll
    for (int j = 0; j < 8; j++) acc[j] = 0.0f;
    acc = wmma16x16_k64(&Els[mt * 16][0], 65, &Wls[0][wave * 16], 65, acc, lane);
    float bb = b1[wave * 16 + l16];
#pragma unroll
    for (int j = 0; j < 8; j++)
      Gls[mt * 16 + j + 8 * half][wave * 16 + l16] = tanhf(acc[j] + bb);
  }
  __syncthreads();

  if (tid < Lc) {
    float s = b2[0];
    for (int d = 0; d < Dd; d++) s += Gls[tid][d] * w2[d];
    Sl[tid] = s;
  }
  __syncthreads();
  if (tid == 0) {
    float m = -3.4e38f; int any = 0;
    for (int l = 0; l < Lc; l++) if (mk[l]) { any = 1; m = fmaxf(m, Sl[l]); }
    float sum = 0.0f;
    for (int l = 0; l < Lc; l++) {
      float e = (any && mk[l]) ? expf(Sl[l] - m) : 0.0f;
      alpha[l] = e; sum += e;
    }
    float inv = any ? (1.0f / sum) : 0.0f;
    for (int l = 0; l < Lc; l++) alpha[l] *= inv;
  }
  __syncthreads();
  if (tid < Dd) {
    float v = 0.0f;
    for (int l = 0; l < Lc; l++) v += alpha[l] * Els[l][tid];
    v_all[bt * Dd + tid] = v;
  }
}

// ---------------- visit-level weights ----------------
__global__ __launch_bounds__(64)
void k_visit(const float* __restrict__ v_all, const int* __restrict__ lengths,
             const unsigned char* __restrict__ hist_visit_mask,
             float* __restrict__ v_cur, float* __restrict__ c_visit)
{
  __shared__ float vc[Dd];
  __shared__ float sc[Hvis];
  const int b = blockIdx.x, tid = threadIdx.x;
  int idx = lengths[b] - 1; if (idx < 0) idx = 0;
  vc[tid] = v_all[(b * Tv + idx) * Dd + tid];
  v_cur[b * Dd + tid] = vc[tid];
  __syncthreads();
  if (tid < Hvis) {
    float s = 0.0f;
    for (int d = 0; d < Dd; d++) s += v_all[(b * Tv + tid) * Dd + d] * vc[d];
    sc[tid] = s * 0.125f;  // 1/sqrt(64)
  }
  __syncthreads();
  if (tid == 0) {
    float m = -3.4e38f; int any = 0;
    for (int h = 0; h < Hvis; h++)
      if (hist_visit_mask[b * Hvis + h]) { any = 1; m = fmaxf(m, sc[h]); }
    float sum = 0.0f; float e[Hvis];
    for (int h = 0; h < Hvis; h++) {
      e[h] = (any && hist_visit_mask[b * Hvis + h]) ? expf(sc[h] - m) : 0.0f;
      sum += e[h];
    }
    float inv = any ? (1.0f / sum) : 0.0f;
    for (int h = 0; h < Hvis; h++) c_visit[b * Hvis + h] = e[h] * inv;
  }
}

__global__ void k_cinst(const int* __restrict__ hist_vidx,
                        const float* __restrict__ c_visit, float* __restrict__ c_inst)
{
  int i = blockIdx.x * 256 + threadIdx.x;
  if (i >= Bsz * Nh) return;
  int b = i >> 10;
  int v = hist_vidx[i];
  if (v < 0) v = 0; if (v > Hvis - 1) v = Hvis - 1;
  c_inst[i] = c_visit[b * Hvis + v];
}

// ---------------- tiny GEMMs (GCN): C[M,64] = A[M,K] @ B[K,64], optional relu
__global__ __launch_bounds__(64)
void k_gemm_n64(const float* __restrict__ A, const float* __restrict__ B,
                float* __restrict__ C, int K, int relu)
{
  int row = blockIdx.x, d = threadIdx.x;
  float s = 0.0f;
  for (int k = 0; k < K; k++) s += A[row * K + k] * B[k * Dd + d];
  if (relu) s = fmaxf(s, 0.0f);
  C[row * Dd + d] = s;
}

__global__ void k_medplus(const float* __restrict__ med_emb, const float* __restrict__ g_ehr,
                          const float* __restrict__ g_ddi, const float* __restrict__ beta_logit,
                          float* __restrict__ med_plus)
{
  int i = blockIdx.x * 256 + threadIdx.x;
  if (i >= NMED * Dd) return;
  float beta = 1.0f / (1.0f + expf(-beta_logit[0]));
  med_plus[i] = med_emb[i] + g_ehr[i] - beta * g_ddi[i];
}

__global__ __launch_bounds__(64)
void k_h0(const float* __restrict__ v_cur, const float* __restrict__ W_h0,
          const float* __restrict__ b_h0, float* __restrict__ h)
{
  __shared__ float vc[Dd];
  int b = blockIdx.x, d = threadIdx.x;
  vc[d] = v_cur[b * Dd + d];
  __syncthreads();
  float s = b_h0[d];
  for (int k = 0; k < Dd; k++) s += vc[k] * W_h0[k * Dd + d];
  h[b * Dd + d] = tanhf(s);
}

// ---------------- GRU step (WMMA): 16 batch rows per block ----------------
__global__ __launch_bounds__(128)
void k_gru(const float* __restrict__ h_in, float* __restrict__ h_out,
           const int* __restrict__ dec_in, int t,
           const float* __restrict__ med_plus, const float* __restrict__ start_emb,
           const float* __restrict__ W_ih, const float* __restrict__ W_hh,
           const float* __restrict__ b_ih, const float* __restrict__ b_hh)
{
  __shared__ float X[16][65];
  __shared__ float Hs[16][65];
  __shared__ float GI[16][196];
  __shared__ float GH[16][196];
  const int b0 = blockIdx.x * 16;
  const int tid = threadIdx.x;

  for (int i = tid; i < 16 * Dd; i += 128) {
    int r = i >> 6, d = i & 63;
    int b = b0 + r;
    int tok = dec_in[b * MAXLEN + t];
    float x = 0.0f;
    if (tok == 1) x = start_emb[d];
    else if (tok >= 2) {
      int mid = tok - 2; if (mid > NMED - 1) mid = NMED - 1;
      x = med_plus[mid * Dd + d];
    }
    X[r][d]  = x;
    Hs[r][d] = h_in[b * Dd + d];
  }
  __syncthreads();

  const int wave = tid >> 5, lane = tid & 31;
  const int half = lane >> 4, l16 = lane & 15;
  for (int job = wave; job < 24; job += 4) {        // uniform per wave
    const int which = job / 12;                     // 0: gi=X@W_ih^T, 1: gh=H@W_hh^T
    const int nt = job % 12;                        // column tile of 192
    const float* Amat = which ? &Hs[0][0] : &X[0][0];
    const float* Wt   = (which ? W_hh : W_ih) + nt * 16 * Dd;  // rows nt*16.. of W
    v8f acc;
#pragma unroll
    for (int j = 0; j < 8; j++) acc[j] = 0.0f;
    acc = wmma16x16_k64_bt(Amat, 65, Wt, Dd, acc, lane);
    float* Dst = which ? &GH[0][0] : &GI[0][0];
#pragma unroll
    for (int j = 0; j < 8; j++)
      Dst[(j + 8 * half) * 196 + nt * 16 + l16] = acc[j];
  }
  __syncthreads();

  for (int i = tid; i < 16 * Dd; i += 128) {
    int r = i >> 6, d = i & 63;
    int b = b0 + r;
    float gir = GI[r][d]        + b_ih[d],        ghr = GH[r][d]        + b_hh[d];
    float giz = GI[r][Dd + d]   + b_ih[Dd + d],   ghz = GH[r][Dd + d]   + b_hh[Dd + d];
    float gin = GI[r][2*Dd + d] + b_ih[2*Dd + d], ghn = GH[r][2*Dd + d] + b_hh[2*Dd + d];
    float rr = 1.0f / (1.0f + expf(-(gir + ghr)));
    float zz = 1.0f / (1.0f + expf(-(giz + ghz)));
    float nn = tanhf(gin + rr * ghn);
    h_out[b * Dd + d] = (1.0f - zz) * nn + zz * Hs[r][d];
  }
}

// ---------------- generate head: p_g, h_cq, gate ----------------
__global__ __launch_bounds__(192)
void k_head(const float* __restrict__ h, const float* __restrict__ W_gen,
            const float* __restrict__ b_gen, const float* __restrict__ W_cq,
            const float* __restrict__ b_cq, const float* __restrict__ W_gate,
            const float* __restrict__ b_gate, float* __restrict__ p_g,
            float* __restrict__ hcq, float* __restrict__ gate)
{
  __shared__ float hl[Dd];
  __shared__ float lg[NTOK];
  __shared__ float s_m, s_inv;
  const int b = blockIdx.x, tid = threadIdx.x;
  if (tid < Dd) hl[tid] = h[b * Dd + tid];
  __syncthreads();
  if (tid < NTOK) {
    float s = b_gen[tid];
    for (int k = 0; k < Dd; k++) s += hl[k] * W_gen[k * NTOK + tid];
    lg[tid] = (tid == 1) ? -1e9f : s;
  }
  __syncthreads();
  if (tid == 0) {
    float m = -3.4e38f;
    for (int j = 0; j < NTOK; j++) m = fmaxf(m, lg[j]);
    float su = 0.0f;
    for (int j = 0; j < NTOK; j++) su += expf(lg[j] - m);
    s_m = m; s_inv = 1.0f / su;
  }
  __syncthreads();
  if (tid < NTOK) p_g[b * NTOK + tid] = expf(lg[tid] - s_m) * s_inv;
  if (tid < Dd) {
    float s = b_cq[tid];
    for (int k = 0; k < Dd; k++) s += hl[k] * W_cq[k * Dd + tid];
    hcq[b * Dd + tid] = s;
  }
  if (tid == 191) {
    float s = b_gate[0];
    for (int k = 0; k < Dd; k++) s += hl[k] * W_gate[k];
    gate[b] = 1.0f / (1.0f + expf(-s));
  }
}

// ---------------- copy attention + mix + per-sample loss ----------------
__global__ __launch_bounds__(256)
void k_copy(const float* __restrict__ hcq, const float* __restrict__ med_plus,
            const int* __restrict__ hist_tok, const unsigned char* __restrict__ hist_mask,
            const float* __restrict__ c_inst, const float* __restrict__ p_g,
            const float* __restrict__ gate, const int* __restrict__ dec_out, int t,
            float* __restrict__ stepl)
{
  __shared__ float hq[Dd];
  __shared__ float qv[Nh];
  __shared__ float wv[Nh];
  __shared__ int tk[Nh];
  __shared__ unsigned char mks[Nh];
  __shared__ float bins[NTOK];
  __shared__ float sred[256];
  __shared__ float s_m, s_inv; __shared__ int s_any;
  const int b = blockIdx.x, tid = threadIdx.x;
  if (tid < Dd) hq[tid] = hcq[b * Dd + tid];
  __syncthreads();

  for (int n = tid; n < Nh; n += 256) {
    int tok = hist_tok[b * Nh + n];
    unsigned char m = hist_mask[b * Nh + n];
    tk[n] = tok; mks[n] = m;
    float qh = 0.0f;
    if (m && tok >= 2) {
      int mid = tok - 2; if (mid > NMED - 1) mid = NMED - 1;
      float s = 0.0f;
      for (int d = 0; d < Dd; d++) s += hq[d] * med_plus[mid * Dd + d];
      qh = s * 0.125f;
    }
    qv[n] = qh;
  }
  __syncthreads();
  // masked max
  float pm = -3.4e38f;
  for (int n = tid; n < Nh; n += 256) if (mks[n]) pm = fmaxf(pm, qv[n]);
  sred[tid] = pm; __syncthreads();
  for (int s = 128; s > 0; s >>= 1) {
    if (tid < s) sred[tid] = fmaxf(sred[tid], sred[tid + s]);
    __syncthreads();
  }
  if (tid == 0) { s_m = sred[0]; s_any = (sred[0] > -1e37f) ? 1 : 0; }
  __syncthreads();
  float m = s_m; int any = s_any;
  __syncthreads();
  // exp + sum
  float ps = 0.0f;
  for (int n = tid; n < Nh; n += 256) {
    float e = (any && mks[n]) ? expf(qv[n] - m) : 0.0f;
    wv[n] = e; ps += e;
  }
  sred[tid] = ps; __syncthreads();
  for (int s = 128; s > 0; s >>= 1) {
    if (tid < s) sred[tid] += sred[tid + s];
    __syncthreads();
  }
  if (tid == 0) s_inv = any ? (1.0f / sred[0]) : 0.0f;
  __syncthreads();
  float inv = s_inv;
  for (int n = tid; n < Nh; n += 256)
    wv[n] = wv[n] * inv * c_inst[b * Nh + n];   // wv==0 where unmasked
  __syncthreads();
  // deterministic scatter: bin owner scans tokens in fixed order
  if (tid < NTOK) {
    float s = 0.0f;
    if (tid >= 2)
      for (int n = 0; n < Nh; n++) if (tk[n] == tid) s += wv[n];
    bins[tid] = s;
  }
  __syncthreads();
  if (tid == 0) {
    float mass = 0.0f;
    for (int j = 2; j < NTOK; j++) mass += bins[j];
    float wg = (mass > 0.0f) ? gate[b] : 1.0f;
    int tgt = dec_out[b * MAXLEN + t];
    if (tgt < 0) tgt = 0; if (tgt > NTOK - 1) tgt = NTOK - 1;
    float pc = (mass > 0.0f) ? (bins[tgt] / (mass + 1e-12f)) : 0.0f;
    float p = wg * p_g[b * NTOK + tgt] + (1.0f - wg) * pc;
    p = fmaxf(p, 1e-12f);
    stepl[b] = -logf(p);
  }
}

__global__ __launch_bounds__(512)
void k_stepsum(const float* __restrict__ stepl, float* __restrict__ lparts, int t)
{
  __shared__ float r[512];
  int tid = threadIdx.x;
  r[tid] = stepl[tid];
  __syncthreads();
  for (int s = 256; s > 0; s >>= 1) {
    if (tid < s) r[tid] += r[tid + s];
    __syncthreads();
  }
  if (tid == 0) lparts[t] = r[0];
}

__global__ void k_final(const float* __restrict__ lparts, float* __restrict__ out)
{
  float s = 0.0f;
  for (int t = 0; t < MAXLEN; t++) s += lparts[t];
  out[0] = s / (float)(Bsz * MAXLEN);
}

// ---------------- launcher ----------------
extern "C" void kernel_launch(void* const* d_in, const int* in_sizes, int n_in,
                              void* d_out, int out_size, void* d_ws, size_t ws_size,
                              hipStream_t stream) {
  (void)in_sizes; (void)n_in; (void)out_size; (void)ws_size;
  const int*  diag_ids        = (const int*)d_in[0];
  const unsigned char* diag_mask = (const unsigned char*)d_in[1];
  const int*  lengths         = (const int*)d_in[2];
  const unsigned char* hvmask = (const unsigned char*)d_in[3];
  const int*  hist_tok        = (const int*)d_in[4];
  const int*  hist_vidx       = (const int*)d_in[5];
  const unsigned char* hmask  = (const unsigned char*)d_in[6];
  const int*  dec_in          = (const int*)d_in[7];
  const int*  dec_out         = (const int*)d_in[8];
  const float* A_ehr          = (const float*)d_in[9];
  const float* A_ddi          = (const float*)d_in[10];
  const float* diag_emb       = (const float*)d_in[11];
  const float* med_emb        = (const float*)d_in[12];
  const float* W_att1         = (const float*)d_in[13];
  const float* b_att1         = (const float*)d_in[14];
  const float* w_att2         = (const float*)d_in[15];
  const float* b_att2         = (const float*)d_in[16];
  const float* ehr_W1         = (const float*)d_in[17];
  const float* ehr_W2         = (const float*)d_in[18];
  const float* ddi_W1         = (const float*)d_in[19];
  const float* ddi_W2         = (const float*)d_in[20];
  const float* beta_logit     = (const float*)d_in[21];
  const float* W_h0           = (const float*)d_in[22];
  const float* b_h0           = (const float*)d_in[23];
  const float* W_ih           = (const float*)d_in[24];
  const float* W_hh           = (const float*)d_in[25];
  const float* b_ih           = (const float*)d_in[26];
  const float* b_hh           = (const float*)d_in[27];
  const float* start_emb      = (const float*)d_in[28];
  const float* W_gen          = (const float*)d_in[29];
  const float* b_gen          = (const float*)d_in[30];
  const float* W_cq           = (const float*)d_in[31];
  const float* b_cq           = (const float*)d_in[32];
  const float* W_gate         = (const float*)d_in[33];
  const float* b_gate         = (const float*)d_in[34];
  float* out = (float*)d_out;

  float* w = (float*)d_ws;
  float* v_all    = w;                       // 512*16*64
  float* v_cur    = v_all + Bsz*Tv*Dd;       // 512*64
  float* c_visit  = v_cur + Bsz*Dd;          // 512*15
  float* c_inst   = c_visit + Bsz*Hvis;      // 512*1024
  float* med_plus = c_inst + Bsz*Nh;         // 150*64
  float* g_t1     = med_plus + NMED*Dd;      // 150*64
  float* g_t2     = g_t1 + NMED*Dd;          // 150*64
  float* g_ehr    = g_t2 + NMED*Dd;          // 150*64
  float* g_ddi    = g_ehr + NMED*Dd;         // 150*64
  float* hbuf0    = g_ddi + NMED*Dd;         // 512*64
  float* hbuf1    = hbuf0 + Bsz*Dd;          // 512*64
  float* p_g      = hbuf1 + Bsz*Dd;          // 512*152
  float* hcq      = p_g + Bsz*NTOK;          // 512*64
  float* gate     = hcq + Bsz*Dd;            // 512
  float* stepl    = gate + Bsz;              // 512
  float* lparts   = stepl + Bsz;             // 45

  // encode
  k_enc<<<Bsz*Tv, 128, 0, stream>>>(diag_ids, diag_mask, diag_emb, W_att1, b_att1,
                                    w_att2, b_att2, v_all);
  k_visit<<<Bsz, 64, 0, stream>>>(v_all, lengths, hvmask, v_cur, c_visit);
  k_cinst<<<(Bsz*Nh + 255)/256, 256, 0, stream>>>(hist_vidx, c_visit, c_inst);

  // GCNs: g = (A @ relu((A@X)@W1)) @ W2
  k_gemm_n64<<<NMED, 64, 0, stream>>>(A_ehr, med_emb, g_t1, NMED, 0);
  k_gemm_n64<<<NMED, 64, 0, stream>>>(g_t1, ehr_W1, g_t2, Dd, 1);
  k_gemm_n64<<<NMED, 64, 0, stream>>>(A_ehr, g_t2, g_t1, NMED, 0);
  k_gemm_n64<<<NMED, 64, 0, stream>>>(g_t1, ehr_W2, g_ehr, Dd, 0);
  k_gemm_n64<<<NMED, 64, 0, stream>>>(A_ddi, med_emb, g_t1, NMED, 0);
  k_gemm_n64<<<NMED, 64, 0, stream>>>(g_t1, ddi_W1, g_t2, Dd, 1);
  k_gemm_n64<<<NMED, 64, 0, stream>>>(A_ddi, g_t2, g_t1, NMED, 0);
  k_gemm_n64<<<NMED, 64, 0, stream>>>(g_t1, ddi_W2, g_ddi, Dd, 0);
  k_medplus<<<(NMED*Dd + 255)/256, 256, 0, stream>>>(med_emb, g_ehr, g_ddi,
                                                     beta_logit, med_plus);
  // h0
  k_h0<<<Bsz, 64, 0, stream>>>(v_cur, W_h0, b_h0, hbuf0);

  // decode
  float* hin = hbuf0; float* hout = hbuf1;
  for (int t = 0; t < MAXLEN; t++) {
    k_gru<<<Bsz/16, 128, 0, stream>>>(hin, hout, dec_in, t, med_plus, start_emb,
                                      W_ih, W_hh, b_ih, b_hh);
    k_head<<<Bsz, 192, 0, stream>>>(hout, W_gen, b_gen, W_cq, b_cq, W_gate, b_gate,
                                    p_g, hcq, gate);
    k_copy<<<Bsz, 256, 0, stream>>>(hcq, med_plus, hist_tok, hmask, c_inst, p_g,
                                    gate, dec_out, t, stepl);
    k_stepsum<<<1, 512, 0, stream>>>(stepl, lparts, t);
    float* tmp = hin; hin = hout; hout = tmp;
  }
  k_final<<<1, 1, 0, stream>>>(lparts, out);
}